// STDPLinear_12360915878899
// MI455X (gfx1250) — compile-verified
//
#include <hip/hip_runtime.h>

// ---------------------------------------------------------------------------
// STDP Linear layer for MI455X (gfx1250, wave32, WMMA).
//
// Roofline: GEMM is 1.07 GFLOP (negligible). Output traces are 2 x 64 MB
// streaming writes + 128 MB reads of initial traces => ~256 MB ~ 11 us at
// 23.3 TB/s. So we use closed-form linear-recurrence collapse for the traces
// and spend the bandwidth only on the unavoidable [B,OUT,IN] broadcasts.
// GEMM uses V_WMMA_F32_16X16X4_F32 (exact fp32) because spikes are threshold
// decisions and gate everything downstream.
// ---------------------------------------------------------------------------

typedef __attribute__((ext_vector_type(2))) float v2f;
typedef __attribute__((ext_vector_type(4))) float v4f;
typedef __attribute__((ext_vector_type(8))) float v8f;

#define T_STEPS 32
#define BATCH   16
#define IN_F    1024
#define OUT_F   1024
#define M_DIM   (T_STEPS * BATCH)      // 512 GEMM rows
#define BO      (BATCH * OUT_F)        // 16384
#define BI      (BATCH * IN_F)         // 16384

// ---------------------------------------------------------------------------
// Kernel 1: psp[m][n] = sum_k x[m][k] * w[n][k]   (m = t*B+b, n = out feature)
// One wave -> 16(M) x 64(N) strip: 4 f32 WMMA accumulators, A-frag reused 4x.
// 512 waves total = 64 blocks x 8 waves.
// ---------------------------------------------------------------------------
__global__ void __launch_bounds__(256)
stdp_psp_gemm_f32wmma(const float* __restrict__ x,
                      const float* __restrict__ w,
                      float* __restrict__ psp)
{
    const int lane = threadIdx.x & 31;
    const int wave = (blockIdx.x << 3) + (threadIdx.x >> 5);   // 0..511
    const int m0 = (wave & 31) << 4;    // 32 M-tiles of 16
    const int n0 = (wave >> 5) << 6;    // 16 N-strips of 64

    // A(16x4 f32) fragment: lane = (k>=2)*16 + m ; vgpr = k&1
    // B(4x16 f32) fragment: lane = (k>=2)*16 + n ; vgpr = k&1
    const int r  = lane & 15;          // m for A, n for B
    const int kh = (lane >> 4) << 1;   // 0 or 2: which k-pair this half-wave owns

    const float* aP  = x + (size_t)(m0 + r) * IN_F + kh;
    const float* bP0 = w + (size_t)(n0 +  0 + r) * IN_F + kh;
    const float* bP1 = w + (size_t)(n0 + 16 + r) * IN_F + kh;
    const float* bP2 = w + (size_t)(n0 + 32 + r) * IN_F + kh;
    const float* bP3 = w + (size_t)(n0 + 48 + r) * IN_F + kh;

    v8f acc0 = {}, acc1 = {}, acc2 = {}, acc3 = {};

    #pragma unroll 4
    for (int k0 = 0; k0 < IN_F; k0 += 4) {
        v2f a  = *(const v2f*)(aP  + k0);
        v2f b0 = *(const v2f*)(bP0 + k0);
        v2f b1 = *(const v2f*)(bP1 + k0);
        v2f b2 = *(const v2f*)(bP2 + k0);
        v2f b3 = *(const v2f*)(bP3 + k0);
        // 8 args: (neg_a, A, neg_b, B, c_mod, C, reuse_a, reuse_b)
        acc0 = __builtin_amdgcn_wmma_f32_16x16x4_f32(false, a, false, b0,
                                                     (short)0, acc0, false, false);
        acc1 = __builtin_amdgcn_wmma_f32_16x16x4_f32(false, a, false, b1,
                                                     (short)0, acc1, false, false);
        acc2 = __builtin_amdgcn_wmma_f32_16x16x4_f32(false, a, false, b2,
                                                     (short)0, acc2, false, false);
        acc3 = __builtin_amdgcn_wmma_f32_16x16x4_f32(false, a, false, b3,
                                                     (short)0, acc3, false, false);
    }

    // D(16x16 f32): vgpr v holds row m0 + (lane>=16 ? 8:0) + v, col = lane&15
    const int n  = lane & 15;
    const int mh = (lane >> 4) << 3;   // 0 or 8
    #pragma unroll
    for (int v = 0; v < 8; ++v) {
        float* row = psp + (size_t)(m0 + mh + v) * OUT_F + n0 + n;
        row[0]  = acc0[v];
        row[16] = acc1[v];
        row[32] = acc2[v];
        row[48] = acc3[v];
    }
}

// ---------------------------------------------------------------------------
// Kernel 2: LIF soma scan over T per (b,o) + collapsed output-trace vector.
// u' = 0.5*u + 0.5*p ; spike = (u' >= 1) ; hard reset to 0.
// otrv[b,o] = sum_t 0.5^(31-t) * spike_t   (v = 0.5*v + s recurrence)
// psp layout [t*16+b][o] == out[t,b,o] flat layout, so writes are direct.
// ---------------------------------------------------------------------------
__global__ void __launch_bounds__(256)
stdp_lif_scan(const float* __restrict__ psp,
              float* __restrict__ out,
              float* __restrict__ otrv)
{
    const int idx = blockIdx.x * 256 + threadIdx.x;   // 0..BO-1
    float u = 0.0f, v = 0.0f;
    #pragma unroll
    for (int t = 0; t < T_STEPS; ++t) {
        const float p = psp[(size_t)t * BO + idx];
        u = 0.5f * u + 0.5f * p;
        const float s = (u >= 1.0f) ? 1.0f : 0.0f;
        out[(size_t)t * BO + idx] = s;
        u = (s != 0.0f) ? 0.0f : u;   // hard reset, U_REST = 0
        v = 0.5f * v + s;
    }
    otrv[idx] = v;
}

// ---------------------------------------------------------------------------
// Kernel 3: collapsed input-trace vector xtr[b,i] = sum_t 0.5^(31-t) x[t,b,i]
// ---------------------------------------------------------------------------
__global__ void __launch_bounds__(256)
stdp_xtrace(const float* __restrict__ x, float* __restrict__ xtr)
{
    const int idx = blockIdx.x * 256 + threadIdx.x;   // 0..BI-1
    float v = 0.0f;
    #pragma unroll
    for (int t = 0; t < T_STEPS; ++t)
        v = 0.5f * v + x[(size_t)t * BI + idx];
    xtr[idx] = v;
}

// ---------------------------------------------------------------------------
// Kernel 4: bandwidth kernel. Broadcast trace vectors into [B,OUT,IN] with
// the 0.5^32-decayed initial traces. 128-bit ext-vector + non-temporal
// loads/stores: these 2 x 64MB outputs are written once and never re-read.
// ---------------------------------------------------------------------------
__global__ void __launch_bounds__(256)
stdp_trace_expand(const float* __restrict__ itr0,
                  const float* __restrict__ otr0,
                  const float* __restrict__ xtr,
                  const float* __restrict__ otrv,
                  float* __restrict__ itrOut,
                  float* __restrict__ otrOut)
{
    const float C32 = 2.3283064365386963e-10f;        // 0.5^32
    const size_t idx = ((size_t)blockIdx.x * 256 + threadIdx.x) << 2; // elem idx, /4 aligned
    const int b = (int)(idx >> 20);          // / (OUT_F*IN_F)
    const int o = (int)((idx >> 10) & 1023); // / IN_F % OUT_F
    const int i = (int)(idx & 1023);

    const v4f  xv = *(const v4f*)(xtr + b * IN_F + i);  // 4 consecutive i
    const float ov = otrv[b * OUT_F + o];               // same o for all 4

    const v4f it0 = __builtin_nontemporal_load((const v4f*)(itr0 + idx));
    const v4f ot0 = __builtin_nontemporal_load((const v4f*)(otr0 + idx));

    v4f ri = C32 * it0 + xv;
    v4f ro = C32 * ot0 + (v4f){ov, ov, ov, ov};

    __builtin_nontemporal_store(ri, (v4f*)(itrOut + idx));
    __builtin_nontemporal_store(ro, (v4f*)(otrOut + idx));
}

// ---------------------------------------------------------------------------
extern "C" void kernel_launch(void* const* d_in, const int* in_sizes, int n_in,
                              void* d_out, int out_size, void* d_ws, size_t ws_size,
                              hipStream_t stream) {
    const float* x    = (const float*)d_in[0];   // [T,B,IN]
    const float* w    = (const float*)d_in[1];   // [OUT,IN]
    const float* itr0 = (const float*)d_in[2];   // [B,OUT,IN]
    const float* otr0 = (const float*)d_in[3];   // [B,OUT,IN]

    float* out    = (float*)d_out;                                // [T,B,OUT]
    float* itrOut = out + (size_t)T_STEPS * BO;                   // [B,OUT,IN]
    float* otrOut = itrOut + (size_t)BATCH * OUT_F * IN_F;        // [B,OUT,IN]

    char* ws   = (char*)d_ws;
    float* psp  = (float*)ws;                                     // 512*1024*4 = 2 MB
    float* xtr  = (float*)(ws + (size_t)M_DIM * OUT_F * 4);       // 64 KB
    float* otrv = xtr + BI;                                       // 64 KB

    // 1) fp32 WMMA GEMM: 512 waves (64 blocks x 8 waves)
    stdp_psp_gemm_f32wmma<<<64, 256, 0, stream>>>(x, w, psp);
    // 2) LIF scan + output-trace collapse: 16384 threads
    stdp_lif_scan<<<BO / 256, 256, 0, stream>>>(psp, out, otrv);
    // 3) input-trace collapse: 16384 threads
    stdp_xtrace<<<BI / 256, 256, 0, stream>>>(x, xtr);
    // 4) streaming broadcast of both [B,OUT,IN] traces (v4f per thread)
    stdp_trace_expand<<<(BATCH * OUT_F * IN_F) / (256 * 4), 256, 0, stream>>>(
        itr0, otr0, xtr, otrv, itrOut, otrOut);
}